// GraphConvPoolModule_29094108463623
// MI455X (gfx1250) — compile-verified
//
#include <hip/hip_runtime.h>
#include <hip/hip_bf16.h>

#define N_NODES 100000
#define E_EDGES 1600000
#define D 128
#define K_TOP 50000
#define M_SORT 131072  // next pow2 >= N_NODES

typedef __attribute__((ext_vector_type(2))) float v2f;
typedef __attribute__((ext_vector_type(8))) float v8f;

// ---------------- init: deg=1 (self loop), new_id=-1 ----------------
__global__ void k_init(float* deg, int* new_id) {
  int i = blockIdx.x * blockDim.x + threadIdx.x;
  if (i < N_NODES) { deg[i] = 1.0f; new_id[i] = -1; }
}

// ---------------- degree count over dst ----------------
__global__ void k_deg(const int* __restrict__ dst, float* deg) {
  int e = blockIdx.x * blockDim.x + threadIdx.x;
  if (e < E_EDGES) unsafeAtomicAdd(&deg[dst[e]], 1.0f);
}

__global__ void k_rsqrt(float* deg_inv) {
  int i = blockIdx.x * blockDim.x + threadIdx.x;
  if (i < N_NODES) deg_inv[i] = __frsqrt_rn(deg_inv[i]);
}

// ---------------- h1 = x @ W1 via V_WMMA_F32_16X16X4_F32 ----------------
// one wave -> one 16x16 tile of C; 8 waves/block cover the 8 column tiles.
__global__ void __launch_bounds__(256) k_gemm_wmma(const float* __restrict__ A,
                                                   const float* __restrict__ B,
                                                   float* __restrict__ C) {
  const int wave = threadIdx.x >> 5;   // 0..7: column tile
  const int lane = threadIdx.x & 31;
  const int half = lane >> 4;          // 0: lanes 0-15, 1: lanes 16-31
  const int lr   = lane & 15;
  const int row0 = blockIdx.x * 16;    // 6250 blocks * 16 = 100000 rows exactly
  const int col0 = wave * 16;

  v8f c = {};
  const float* __restrict__ arow = A + (long long)(row0 + lr) * D;
  const float* __restrict__ bcol = B + col0 + lr;
  #pragma unroll 4
  for (int k0 = 0; k0 < D; k0 += 4) {
    const int ka = k0 + 2 * half;      // lanes 0-15: K=k0,k0+1 ; lanes 16-31: K=k0+2,k0+3
    v2f a, b;
    a.x = arow[ka];
    a.y = arow[ka + 1];
    b.x = bcol[(long long)ka * D];
    b.y = bcol[(long long)(ka + 1) * D];
    c = __builtin_amdgcn_wmma_f32_16x16x4_f32(false, a, false, b,
                                              (short)0, c, false, false);
  }
  // C/D layout: VGPR r -> M = row0 + r + 8*half, N = col0 + lr
  float* crow = C + (long long)(row0 + 8 * half) * D + col0 + lr;
  #pragma unroll
  for (int r = 0; r < 8; ++r) crow[(long long)r * D] = c[r];
}

// ---------------- acc = h1 * inv^2 (self-loop term) ----------------
__global__ void k_selfloop(const float* __restrict__ h1,
                           const float* __restrict__ inv, float* acc) {
  int gid = blockIdx.x * blockDim.x + threadIdx.x;  // N*32 threads
  int n = gid >> 5, lane = gid & 31;
  if (n < N_NODES) {
    float s = inv[n] * inv[n];
    float4 v = ((const float4*)(h1 + (long long)n * D))[lane];
    float4 o = make_float4(v.x * s, v.y * s, v.z * s, v.w * s);
    ((float4*)(acc + (long long)n * D))[lane] = o;
  }
}

// ---------------- edge scatter: acc[dst] += h1[src]*norm (1 wave/edge) ----------------
__global__ void k_scatter_feat(const int* __restrict__ src, const int* __restrict__ dst,
                               const float* __restrict__ h1, const float* __restrict__ inv,
                               float* acc) {
  long long gid = (long long)blockIdx.x * blockDim.x + threadIdx.x;  // E*32 threads
  int e = (int)(gid >> 5), lane = (int)(gid & 31);
  if (e < E_EDGES) {
    int s = src[e], d = dst[e];
    float nrm = inv[s] * inv[d];
    const float* __restrict__ hs = h1 + (long long)s * D;
    float* ad = acc + (long long)d * D;
    #pragma unroll
    for (int j = 0; j < 4; ++j) {
      int col = lane + j * 32;  // coalesced across the wave
      unsafeAtomicAdd(&ad[col], hs[col] * nrm);
    }
  }
}

// ---------------- x_conv = relu(acc+b1); s = x_conv . W2 (1 wave/node) ----------------
__global__ void k_relu_dot(float* acc, const float* __restrict__ b1,
                           const float* __restrict__ W2, float* sarr) {
  int gid = blockIdx.x * blockDim.x + threadIdx.x;  // N*32 threads
  int n = gid >> 5, lane = gid & 31;
  if (n >= N_NODES) return;
  float* row = acc + (long long)n * D;
  float part = 0.0f;
  #pragma unroll
  for (int j = 0; j < 4; ++j) {
    int col = lane + j * 32;
    float v = row[col] + b1[col];
    v = v > 0.0f ? v : 0.0f;
    row[col] = v;               // acc becomes x_conv in place
    part = fmaf(v, W2[col], part);
  }
  for (int off = 16; off > 0; off >>= 1) part += __shfl_xor(part, off, 32);
  if (lane == 0) sarr[n] = part;
}

__global__ void k_att_init(const float* __restrict__ sarr, const float* __restrict__ inv,
                           float* att_acc) {
  int n = blockIdx.x * blockDim.x + threadIdx.x;
  if (n < N_NODES) att_acc[n] = sarr[n] * inv[n] * inv[n];
}

__global__ void k_scatter_scalar(const int* __restrict__ src, const int* __restrict__ dst,
                                 const float* __restrict__ sarr, const float* __restrict__ inv,
                                 float* att_acc) {
  int e = blockIdx.x * blockDim.x + threadIdx.x;
  if (e < E_EDGES) {
    int s = src[e], d = dst[e];
    unsafeAtomicAdd(&att_acc[d], sarr[s] * inv[s] * inv[d]);
  }
}

__global__ void k_sigmoid(const float* __restrict__ att_acc, const float* __restrict__ b2,
                          float* att) {
  int n = blockIdx.x * blockDim.x + threadIdx.x;
  if (n < N_NODES) att[n] = 1.0f / (1.0f + __expf(-(att_acc[n] + b2[0])));
}

// ---------------- top-K via bitonic sort of packed u64 keys ----------------
__device__ inline unsigned int f2ord(float f) {
  unsigned int u = __float_as_uint(f);
  return (u & 0x80000000u) ? ~u : (u | 0x80000000u);  // larger float -> larger uint
}

__global__ void k_build_keys(const float* __restrict__ att, unsigned long long* keys) {
  int i = blockIdx.x * blockDim.x + threadIdx.x;
  if (i < M_SORT) {
    if (i < N_NODES)
      keys[i] = ((unsigned long long)f2ord(att[i]) << 32) |
                (unsigned long long)(~(unsigned int)i);     // ties -> smaller idx first
    else
      keys[i] = 0ull;                                       // pads sort last (scores>0)
  }
}

__global__ void k_bitonic(unsigned long long* keys, int j, int k) {
  int i = blockIdx.x * blockDim.x + threadIdx.x;
  int ixj = i ^ j;
  if (i < M_SORT && ixj > i) {
    unsigned long long a = keys[i], b = keys[ixj];
    bool desc = ((i & k) == 0);   // overall descending sort
    if (desc ? (a < b) : (a > b)) { keys[i] = b; keys[ixj] = a; }
  }
}

__global__ void k_topk_scatter(const unsigned long long* __restrict__ keys, int* new_id) {
  int i = blockIdx.x * blockDim.x + threadIdx.x;
  if (i < K_TOP) {
    int idx = (int)(~(unsigned int)(keys[i] & 0xFFFFFFFFull));
    new_id[idx] = i;
  }
}

__global__ void k_gather_pooled(const unsigned long long* __restrict__ keys,
                                const float* __restrict__ xconv,
                                const float* __restrict__ att, float* out) {
  int gid = blockIdx.x * blockDim.x + threadIdx.x;  // K*32 threads
  int i = gid >> 5, lane = gid & 31;
  if (i < K_TOP) {
    int idx = (int)(~(unsigned int)(keys[i] & 0xFFFFFFFFull));
    float a = att[idx];
    float4 v = ((const float4*)(xconv + (long long)idx * D))[lane];
    float4 o = make_float4(v.x * a, v.y * a, v.z * a, v.w * a);
    ((float4*)(out + (long long)i * D))[lane] = o;
  }
}

__global__ void k_edge_out(const int* __restrict__ src, const int* __restrict__ dst,
                           const int* __restrict__ new_id, int* outE) {
  int e = blockIdx.x * blockDim.x + threadIdx.x;
  if (e < E_EDGES) {
    int ns = new_id[src[e]], nd = new_id[dst[e]];
    bool valid = (ns >= 0) && (nd >= 0);
    outE[e]           = valid ? ns : -1;
    outE[E_EDGES + e] = valid ? nd : -1;
  }
}

// ---------------------------------------------------------------------------
extern "C" void kernel_launch(void* const* d_in, const int* in_sizes, int n_in,
                              void* d_out, int out_size, void* d_ws, size_t ws_size,
                              hipStream_t stream) {
  const float* x  = (const float*)d_in[0];
  const int*   ei = (const int*)d_in[1];     // [2, E] int32: row0 = src, row1 = dst
  const float* W1 = (const float*)d_in[2];
  const float* b1 = (const float*)d_in[3];
  const float* W2 = (const float*)d_in[4];
  const float* b2 = (const float*)d_in[5];
  const int* src = ei;
  const int* dst = ei + E_EDGES;

  char* ws = (char*)d_ws;
  size_t off = 0;
  auto alloc = [&](size_t bytes) { void* p = ws + off; off = (off + bytes + 255) & ~(size_t)255; return p; };
  float* h1      = (float*)alloc((size_t)N_NODES * D * 4);  // x @ W1
  float* acc     = (float*)alloc((size_t)N_NODES * D * 4);  // conv accum -> x_conv
  float* inv     = (float*)alloc((size_t)N_NODES * 4);      // deg -> inv_sqrt (in place)
  float* sarr    = (float*)alloc((size_t)N_NODES * 4);      // x_conv . W2
  float* att_acc = (float*)alloc((size_t)N_NODES * 4);
  float* att     = (float*)alloc((size_t)N_NODES * 4);
  unsigned long long* keys = (unsigned long long*)alloc((size_t)M_SORT * 8);
  int* new_id    = (int*)alloc((size_t)N_NODES * 4);

  float* out_px = (float*)d_out;                       // K_TOP * 128 floats
  int*   out_ei = (int*)((float*)d_out + (size_t)K_TOP * D);  // 2*E int32

  const int B = 256;
  k_init<<<(N_NODES + B - 1) / B, B, 0, stream>>>(inv, new_id);
  k_deg<<<(E_EDGES + B - 1) / B, B, 0, stream>>>(dst, inv);
  k_rsqrt<<<(N_NODES + B - 1) / B, B, 0, stream>>>(inv);

  k_gemm_wmma<<<N_NODES / 16, B, 0, stream>>>(x, W1, h1);

  k_selfloop<<<(N_NODES * 32) / B, B, 0, stream>>>(h1, inv, acc);
  k_scatter_feat<<<(int)(((long long)E_EDGES * 32) / B), B, 0, stream>>>(src, dst, h1, inv, acc);
  k_relu_dot<<<(N_NODES * 32) / B, B, 0, stream>>>(acc, b1, W2, sarr);

  k_att_init<<<(N_NODES + B - 1) / B, B, 0, stream>>>(sarr, inv, att_acc);
  k_scatter_scalar<<<(E_EDGES + B - 1) / B, B, 0, stream>>>(src, dst, sarr, inv, att_acc);
  k_sigmoid<<<(N_NODES + B - 1) / B, B, 0, stream>>>(att_acc, b2, att);

  k_build_keys<<<M_SORT / B, B, 0, stream>>>(att, keys);
  for (int k = 2; k <= M_SORT; k <<= 1)
    for (int j = k >> 1; j > 0; j >>= 1)
      k_bitonic<<<M_SORT / B, B, 0, stream>>>(keys, j, k);

  k_topk_scatter<<<(K_TOP + B - 1) / B, B, 0, stream>>>(keys, new_id);
  k_gather_pooled<<<(K_TOP * 32) / B, B, 0, stream>>>(keys, acc, att, out_px);
  k_edge_out<<<(E_EDGES + B - 1) / B, B, 0, stream>>>(src, dst, new_id, out_ei);
}